// ReluField_44014824849656
// MI455X (gfx1250) — compile-verified
//
#include <hip/hip_runtime.h>

#define BLOCK 256
#define DIM   256

typedef __attribute__((ext_vector_type(3))) float v3f;
typedef __attribute__((ext_vector_type(4))) float v4f;

// Trilinear-style gather + weighted sum + ReLU over a [256,256,256,4] f32 field.
// Memory-bound (~312 MB @ 23.3 TB/s -> ~13us floor). CDNA5-specific structure:
//  - 8 corner fetches per point via global_load_async_to_lds_b128 (ASYNCcnt),
//    staged corner-major in LDS for conflict-free ds_load_b128 readback.
//  - Streaming data (coords in, result out) tagged TH_*_NT so the 192MB L2
//    is reserved for the 256MB field (gathers keep default RT hint).
__global__ __launch_bounds__(BLOCK) void ReluField_kernel(
    const float* __restrict__ coords,   // [N,3]
    const float* __restrict__ field,    // [256,256,256,4]
    float* __restrict__ out,            // [N,4]
    int n)
{
    // Corner-major stage: consecutive lanes own consecutive 16B chunks.
    __shared__ __align__(16) float stage[8][BLOCK][4];   // 32 KB

    const int tid = threadIdx.x;
    const int gi  = blockIdx.x * BLOCK + tid;
    // Keep EXEC full everywhere: clamp the tail lanes onto point n-1.
    // They recompute the identical value, so their duplicate store is benign.
    const int i   = gi < n ? gi : (n - 1);

    // 12B coordinate load, non-temporal (read-once stream; keep out of L2).
    const unsigned long long caddr =
        (unsigned long long)reinterpret_cast<const char*>(coords + 3 * i);
    v3f c;
    asm volatile("global_load_b96 %0, %1, off th:TH_LOAD_NT"
                 : "=v"(c) : "v"(caddr) : "memory");
    // Hand-issued VMEM load is not compiler-tracked: tie the result through the
    // wait so no use can be scheduled into the latency window.
    asm volatile("s_wait_loadcnt 0x0" : "+v"(c) :: "memory");

    const float xs = c.x * (float)DIM;
    const float ys = c.y * (float)DIM;
    const float zs = c.z * (float)DIM;
    const float flx = floorf(xs), fly = floorf(ys), flz = floorf(zs);
    const float fx = xs - flx, fy = ys - fly, fz = zs - flz;

    const int ix0 = (int)flx, iy0 = (int)fly, iz0 = (int)flz;
    const int ix[2] = { min(max(ix0, 0), DIM - 1), min(max(ix0 + 1, 0), DIM - 1) };
    const int iy[2] = { min(max(iy0, 0), DIM - 1), min(max(iy0 + 1, 0), DIM - 1) };
    const int iz[2] = { min(max(iz0, 0), DIM - 1), min(max(iz0 + 1, 0), DIM - 1) };
    // Reference weight: prod_d |(floor+offset)_d - x_d|  (unclamped corner coordinate)
    const float wx[2] = { fx, 1.0f - fx };
    const float wy[2] = { fy, 1.0f - fy };
    const float wz[2] = { fz, 1.0f - fz };

    // Issue the 8 corner gathers (16B each: all 4 channels) asynchronously into LDS.
    #pragma unroll
    for (int k = 0; k < 8; ++k) {
        const int bx = (k >> 2) & 1;
        const int by = (k >> 1) & 1;
        const int bz = k & 1;
        const int idx = ((ix[bx] * DIM + iy[by]) * DIM + iz[bz]) * 4;   // < 2^27, int ok
        const unsigned long long gaddr =
            (unsigned long long)reinterpret_cast<const char*>(field + idx);
        // Low 32 bits of a generic shared pointer == LDS byte address (LDS aperture rule).
        const unsigned laddr =
            (unsigned)(unsigned long long)reinterpret_cast<const char*>(&stage[k][tid][0]);
        asm volatile("global_load_async_to_lds_b128 %0, %1, off"
                     :
                     : "v"(laddr), "v"(gaddr)
                     : "memory");
    }

    // Wait for all 8 async gathers of this wave.
    asm volatile("s_wait_asynccnt 0x0" ::: "memory");

    float ax = 0.0f, ay = 0.0f, az = 0.0f, aw = 0.0f;
    #pragma unroll
    for (int k = 0; k < 8; ++k) {
        const int bx = (k >> 2) & 1;
        const int by = (k >> 1) & 1;
        const int bz = k & 1;
        const float w = wx[bx] * wy[by] * wz[bz];
        const float4 v = *reinterpret_cast<const float4*>(&stage[k][tid][0]);  // ds_load_b128
        ax = fmaf(v.x, w, ax);
        ay = fmaf(v.y, w, ay);
        az = fmaf(v.z, w, az);
        aw = fmaf(v.w, w, aw);
    }

    v4f r;
    r.x = fmaxf(ax, 0.0f);
    r.y = fmaxf(ay, 0.0f);
    r.z = fmaxf(az, 0.0f);
    r.w = fmaxf(aw, 0.0f);

    // Non-temporal 16B result store (write-once stream; keep out of L2).
    const unsigned long long oaddr =
        (unsigned long long)reinterpret_cast<char*>(out + 4 * (long long)i);
    asm volatile("global_store_b128 %0, %1, off th:TH_STORE_NT"
                 :
                 : "v"(oaddr), "v"(r)
                 : "memory");
}

extern "C" void kernel_launch(void* const* d_in, const int* in_sizes, int n_in,
                              void* d_out, int out_size, void* d_ws, size_t ws_size,
                              hipStream_t stream) {
    const float* coords = (const float*)d_in[0];   // [N,3] f32
    const float* field  = (const float*)d_in[1];   // [256,256,256,4] f32
    float* out = (float*)d_out;                    // [N,4] f32

    const int n = in_sizes[0] / 3;
    const int blocks = (n + BLOCK - 1) / BLOCK;
    ReluField_kernel<<<blocks, BLOCK, 0, stream>>>(coords, field, out, n);
}